// SSDLoss_3796751089676
// MI455X (gfx1250) — compile-verified
//
#include <hip/hip_runtime.h>
#include <hip/hip_bf16.h>
#include <stdint.h>

// ---------------- problem constants ----------------
#define B_      64
#define A_      24564
#define C_      81
#define ROW_    85                    // C + 4 box coords
#define N_      (B_ * A_)             // 1,572,096 anchors
#define ALPHA_  1.0f
#define RATIO_  3.0f

// ---------------- tiling ----------------
#define TILE_    64                   // anchors per block tile (one thread per anchor)
#define TILE_F   (TILE_ * ROW_)       // 5440 floats per tensor per tile
#define TILE_V4  (TILE_F / 4)         // 1360 float4 (64*85 % 4 == 0 -> 16B aligned tiles)
#define NTILES_  (N_ / TILE_)         // 24564 (exact)
#define GRID1_   1024                 // blocks for main pass
#define GRIDH_   512                  // blocks for histogram / final sum passes

// async pipeline: every wave issues exactly this many async b128 ops per tile
#define ASYNC_ITERS     22            // ceil(1360/64)
#define ASYNC_PER_TILE  (2 * ASYNC_ITERS)   // 44 (both tensors)

// dynamic LDS: two buffers, each holding yt[5440] + yp[5440]
#define BUF_F       (2 * TILE_F)              // 10880 floats per buffer
#define SMEM_FLOATS (2 * BUF_F)               // 21760 floats
#define SMEM_BYTES  (SMEM_FLOATS * 4)         // 87,040 B (< 320KB/WGP on CDNA5)

// ---------------- workspace layout (bytes) ----------------
#define KEYS_OFF   0
#define STATE_OFF  (N_ * 4)                     // uint[16]
#define SCAL_OFF   (STATE_OFF + 64)             // float[16]
#define HIST_OFF   (SCAL_OFF + 64)              // uint[256]
#define PPOS_OFF   (HIST_OFF + 1024)            // float[GRID1_]
#define PLOC_OFF   (PPOS_OFF + GRID1_ * 4)      // float[GRID1_]
#define PNP_OFF    (PLOC_OFF + GRID1_ * 4)      // uint [GRID1_]
#define PNN_OFF    (PNP_OFF + GRID1_ * 4)       // uint [GRID1_]
#define PSUM_OFF   (PNN_OFF + GRID1_ * 4)       // float[GRIDH_]
#define PCNT_OFF   (PSUM_OFF + GRIDH_ * 4)      // uint [GRIDH_]

// state indices
#define S_PREFIX 0
#define S_KREM   1
#define S_KTOT   2
#define S_NPOS   3
#define S_NNEG   4

// ---------------- CDNA5 async global->LDS path ----------------
#if defined(__HIP_DEVICE_COMPILE__) && defined(__gfx1250__)
#  if __has_builtin(__builtin_amdgcn_global_load_async_to_lds_b128) && \
      __has_builtin(__builtin_amdgcn_s_wait_asynccnt)
#    define USE_ASYNC 1
#  endif
#endif
#ifndef USE_ASYNC
#  define USE_ASYNC 0
#endif

#if USE_ASYNC
typedef int v4i_t __attribute__((vector_size(16)));
typedef __attribute__((address_space(1))) v4i_t* gv4_ptr_t;   // global src
typedef __attribute__((address_space(3))) v4i_t* lv4_ptr_t;   // LDS dst
#endif

// order-preserving float<->uint key transform (larger float => larger key; key 0 reserved)
__device__ __forceinline__ unsigned f2ord(float f) {
  unsigned u = __float_as_uint(f);
  return (u & 0x80000000u) ? ~u : (u | 0x80000000u);
}
__device__ __forceinline__ float ord2f(unsigned k) {
  unsigned u = (k & 0x80000000u) ? (k & 0x7fffffffu) : ~k;
  return __uint_as_float(u);
}

// ---------------- kernels ----------------
__global__ __launch_bounds__(256) void ssd_init(unsigned* state, float* scal, unsigned* hist) {
  const int tid = threadIdx.x;
  if (tid < 16) { state[tid] = 0u; scal[tid] = 0.0f; }
  hist[tid] = 0u;
}

// Main streaming pass: double-buffered async-DMA pipeline (global -> LDS) with partial
// s_wait_asynccnt: DMA of tile t+1 overlaps compute of tile t. One thread per anchor.
__global__ __launch_bounds__(TILE_) void ssd_main(const float* __restrict__ yp,
                                                  const float* __restrict__ yt,
                                                  unsigned* __restrict__ keys,
                                                  float* __restrict__ ppos,
                                                  float* __restrict__ ploc,
                                                  unsigned* __restrict__ pnp,
                                                  unsigned* __restrict__ pnn) {
  extern __shared__ float smem[];          // [2][ yt[TILE_F] | yp[TILE_F] ]
  __shared__ float redf[TILE_];
  __shared__ unsigned redu[TILE_];

  const int tid = threadIdx.x;
  const float4* ypv = (const float4*)yp;
  const float4* ytv = (const float4*)yt;

  float accPos = 0.0f, accLoc = 0.0f;
  unsigned accNp = 0u, accNn = 0u;
  const float NEGINF = __uint_as_float(0xff800000u);

#if USE_ASYNC
  // issue one tile's DMA into buffer `buf`; exactly ASYNC_PER_TILE ops per wave
  auto issue_tile = [&](int t, int buf) {
    const size_t b4 = (size_t)t * TILE_V4;
    float4* syt4 = (float4*)(smem + buf * BUF_F);
    float4* syp4 = (float4*)(smem + buf * BUF_F + TILE_F);
#pragma unroll 1
    for (int i = tid; i < TILE_ * ASYNC_ITERS; i += TILE_) {
      const int idx = (i < TILE_V4) ? i : (TILE_V4 - 1);   // uniform op count per wave
      __builtin_amdgcn_global_load_async_to_lds_b128(
          (gv4_ptr_t)(void*)(ytv + b4 + idx), (lv4_ptr_t)(void*)(syt4 + idx), 0, 0);
      __builtin_amdgcn_global_load_async_to_lds_b128(
          (gv4_ptr_t)(void*)(ypv + b4 + idx), (lv4_ptr_t)(void*)(syp4 + idx), 0, 0);
    }
  };
#endif

#if USE_ASYNC
  int cur = 0;
  int t0 = blockIdx.x;
  if (t0 < NTILES_) issue_tile(t0, 0);     // prologue
  for (int t = t0; t < NTILES_; t += gridDim.x) {
    const int nxt = t + (int)gridDim.x;
    if (nxt < NTILES_) {
      issue_tile(nxt, cur ^ 1);            // overlap next tile's DMA with this compute
      __builtin_amdgcn_s_wait_asynccnt(ASYNC_PER_TILE);  // in-order => tile t landed
    } else {
      __builtin_amdgcn_s_wait_asynccnt(0);
    }
    __syncthreads();                        // all waves' DMA for tile t visible
    const float* tr = smem + cur * BUF_F + tid * ROW_;
    const float* pr = smem + cur * BUF_F + TILE_F + tid * ROW_;
#else
  for (int t = blockIdx.x; t < NTILES_; t += gridDim.x) {
    const size_t b4 = (size_t)t * TILE_V4;
    float4* syt4 = (float4*)smem;
    float4* syp4 = (float4*)(smem + TILE_F);
    for (int i = tid; i < TILE_V4; i += TILE_) {
      syt4[i] = ytv[b4 + i];
      syp4[i] = ypv[b4 + i];
    }
    __syncthreads();
    const float* tr = smem + tid * ROW_;
    const float* pr = smem + TILE_F + tid * ROW_;
#endif

    float conf = tr[0] * pr[0];
    float mx = NEGINF;                      // max over classes 1..C-1
#pragma unroll 4
    for (int c = 1; c < C_; ++c) {
      const float tv = tr[c];
      conf = fmaf(tv, pr[c], conf);
      mx = fmaxf(mx, tv);
    }
    conf = -conf;

    const bool neg = (tr[0] != 0.0f);
    const bool pos = (mx != 0.0f);

    float loc = 0.0f;
#pragma unroll
    for (int j = 0; j < 4; ++j) {
      const float d = pr[C_ + j] - tr[C_ + j];
      const float ad = fabsf(d);
      loc += (ad < 1.0f) ? 0.5f * d * d : (ad - 0.5f);
    }

    keys[(size_t)t * TILE_ + tid] = neg ? f2ord(conf) : 0u;
    accNp += pos ? 1u : 0u;
    accNn += neg ? 1u : 0u;
    if (pos) { accPos += conf; accLoc += loc; }
    __syncthreads();                        // buffer free for the DMA after next
#if USE_ASYNC
    cur ^= 1;
#endif
  }

  // deterministic block tree reductions
  redf[tid] = accPos; __syncthreads();
  for (int s = TILE_ / 2; s > 0; s >>= 1) { if (tid < s) redf[tid] += redf[tid + s]; __syncthreads(); }
  if (tid == 0) ppos[blockIdx.x] = redf[0];
  __syncthreads();
  redf[tid] = accLoc; __syncthreads();
  for (int s = TILE_ / 2; s > 0; s >>= 1) { if (tid < s) redf[tid] += redf[tid + s]; __syncthreads(); }
  if (tid == 0) ploc[blockIdx.x] = redf[0];
  __syncthreads();
  redu[tid] = accNp; __syncthreads();
  for (int s = TILE_ / 2; s > 0; s >>= 1) { if (tid < s) redu[tid] += redu[tid + s]; __syncthreads(); }
  if (tid == 0) pnp[blockIdx.x] = redu[0];
  __syncthreads();
  redu[tid] = accNn; __syncthreads();
  for (int s = TILE_ / 2; s > 0; s >>= 1) { if (tid < s) redu[tid] += redu[tid + s]; __syncthreads(); }
  if (tid == 0) pnn[blockIdx.x] = redu[0];
}

__global__ __launch_bounds__(256) void ssd_reduce1(const float* __restrict__ ppos,
                                                   const float* __restrict__ ploc,
                                                   const unsigned* __restrict__ pnp,
                                                   const unsigned* __restrict__ pnn,
                                                   unsigned* state, float* scal) {
  __shared__ float rf[256];
  __shared__ unsigned ru[256];
  const int tid = threadIdx.x;
  float sp = 0.0f, sl = 0.0f; unsigned np = 0u, nn = 0u;
  for (int i = tid; i < GRID1_; i += 256) { sp += ppos[i]; sl += ploc[i]; np += pnp[i]; nn += pnn[i]; }
  rf[tid] = sp; __syncthreads();
  for (int s = 128; s; s >>= 1) { if (tid < s) rf[tid] += rf[tid + s]; __syncthreads(); }
  sp = rf[0]; __syncthreads();
  rf[tid] = sl; __syncthreads();
  for (int s = 128; s; s >>= 1) { if (tid < s) rf[tid] += rf[tid + s]; __syncthreads(); }
  sl = rf[0]; __syncthreads();
  ru[tid] = np; __syncthreads();
  for (int s = 128; s; s >>= 1) { if (tid < s) ru[tid] += ru[tid + s]; __syncthreads(); }
  np = ru[0]; __syncthreads();
  ru[tid] = nn; __syncthreads();
  for (int s = 128; s; s >>= 1) { if (tid < s) ru[tid] += ru[tid + s]; __syncthreads(); }
  nn = ru[0];
  if (tid == 0) {
    scal[0] = sp; scal[1] = sl;
    state[S_NPOS] = np; state[S_NNEG] = nn;
    unsigned kt = (unsigned)(int)(RATIO_ * (float)np);  // matches ref int32 truncation
    if (kt > nn) kt = nn;
    state[S_KTOT] = kt; state[S_KREM] = kt; state[S_PREFIX] = 0u;
  }
}

// one radix-select histogram pass (8 bits) under the current prefix
__global__ __launch_bounds__(256) void ssd_hist(const unsigned* __restrict__ keys,
                                                unsigned* __restrict__ hist,
                                                const unsigned* __restrict__ state,
                                                int shift, unsigned himask) {
  __shared__ unsigned h[256];
  const int tid = threadIdx.x;
  h[tid] = 0u;
  __syncthreads();
  const unsigned pref = state[S_PREFIX] & himask;
  if (state[S_KREM] != 0u) {
    for (unsigned i = blockIdx.x * 256u + tid; i < N_; i += gridDim.x * 256u) {
      const unsigned k = keys[i];
      if (k != 0u && (k & himask) == pref) atomicAdd(&h[(k >> shift) & 255u], 1u);
    }
  }
  __syncthreads();
  if (h[tid]) atomicAdd(&hist[tid], h[tid]);
}

// single-block descending suffix scan: pick the bin holding the k-th largest; reset hist
__global__ __launch_bounds__(256) void ssd_scan(unsigned* hist, unsigned* state, int shift) {
  __shared__ unsigned h[256];
  const int tid = threadIdx.x;
  h[tid] = hist[tid];
  __syncthreads();
  if (tid == 0) {
    const unsigned k = state[S_KREM];
    if (k != 0u) {
      unsigned cum = 0u;
      int b = 255;
      for (; b > 0; --b) {
        const unsigned c = h[b];
        if (cum + c >= k) break;
        cum += c;
      }
      state[S_PREFIX] |= ((unsigned)b) << shift;
      state[S_KREM] = k - cum;
    }
  }
  hist[tid] = 0u;  // ready for next pass
}

// sum + count of negative confs strictly above threshold key
__global__ __launch_bounds__(256) void ssd_negsum(const unsigned* __restrict__ keys,
                                                  const unsigned* __restrict__ state,
                                                  float* __restrict__ psum,
                                                  unsigned* __restrict__ pcnt) {
  __shared__ float rf[256];
  __shared__ unsigned ru[256];
  const int tid = threadIdx.x;
  const unsigned kt = state[S_KTOT];
  const unsigned T = state[S_PREFIX];
  float s = 0.0f; unsigned c = 0u;
  if (kt != 0u) {
    for (unsigned i = blockIdx.x * 256u + tid; i < N_; i += gridDim.x * 256u) {
      const unsigned k = keys[i];
      if (k > T) { s += ord2f(k); ++c; }
    }
  }
  rf[tid] = s; ru[tid] = c; __syncthreads();
  for (int st = 128; st; st >>= 1) {
    if (tid < st) { rf[tid] += rf[tid + st]; ru[tid] += ru[tid + st]; }
    __syncthreads();
  }
  if (tid == 0) { psum[blockIdx.x] = rf[0]; pcnt[blockIdx.x] = ru[0]; }
}

__global__ __launch_bounds__(256) void ssd_final(const float* __restrict__ psum,
                                                 const unsigned* __restrict__ pcnt,
                                                 const unsigned* __restrict__ state,
                                                 const float* __restrict__ scal,
                                                 float* __restrict__ out) {
  __shared__ float rf[256];
  __shared__ unsigned ru[256];
  const int tid = threadIdx.x;
  float s = 0.0f; unsigned c = 0u;
  for (int i = tid; i < GRIDH_; i += 256) { s += psum[i]; c += pcnt[i]; }
  rf[tid] = s; ru[tid] = c; __syncthreads();
  for (int st = 128; st; st >>= 1) {
    if (tid < st) { rf[tid] += rf[tid + st]; ru[tid] += ru[tid + st]; }
    __syncthreads();
  }
  if (tid == 0) {
    float negsum = 0.0f;
    const unsigned kt = state[S_KTOT];
    if (kt != 0u) {
      const float tval = ord2f(state[S_PREFIX]);      // value of k-th largest (tie value)
      negsum = rf[0] + (float)(kt - ru[0]) * tval;    // tie correction
    }
    const unsigned np = state[S_NPOS];
    const float denom = (float)(np > 0u ? np : 1u);
    out[0] = (scal[0] + negsum + ALPHA_ * scal[1]) / denom;
  }
}

// ---------------- launcher ----------------
extern "C" void kernel_launch(void* const* d_in, const int* in_sizes, int n_in,
                              void* d_out, int out_size, void* d_ws, size_t ws_size,
                              hipStream_t stream) {
  (void)in_sizes; (void)n_in; (void)out_size; (void)ws_size;
  const float* yp = (const float*)d_in[0];
  const float* yt = (const float*)d_in[1];
  char* ws = (char*)d_ws;

  unsigned* keys  = (unsigned*)(ws + KEYS_OFF);
  unsigned* state = (unsigned*)(ws + STATE_OFF);
  float*    scal  = (float*)(ws + SCAL_OFF);
  unsigned* hist  = (unsigned*)(ws + HIST_OFF);
  float*    ppos  = (float*)(ws + PPOS_OFF);
  float*    ploc  = (float*)(ws + PLOC_OFF);
  unsigned* pnp   = (unsigned*)(ws + PNP_OFF);
  unsigned* pnn   = (unsigned*)(ws + PNN_OFF);
  float*    psum  = (float*)(ws + PSUM_OFF);
  unsigned* pcnt  = (unsigned*)(ws + PCNT_OFF);
  float*    out   = (float*)d_out;

  hipLaunchKernelGGL(ssd_init, dim3(1), dim3(256), 0, stream, state, scal, hist);
  hipLaunchKernelGGL(ssd_main, dim3(GRID1_), dim3(TILE_), SMEM_BYTES, stream,
                     yp, yt, keys, ppos, ploc, pnp, pnn);
  hipLaunchKernelGGL(ssd_reduce1, dim3(1), dim3(256), 0, stream,
                     ppos, ploc, pnp, pnn, state, scal);

  const int shifts[4]      = {24, 16, 8, 0};
  const unsigned himask[4] = {0x00000000u, 0xFF000000u, 0xFFFF0000u, 0xFFFFFF00u};
  for (int p = 0; p < 4; ++p) {
    hipLaunchKernelGGL(ssd_hist, dim3(GRIDH_), dim3(256), 0, stream,
                       keys, hist, state, shifts[p], himask[p]);
    hipLaunchKernelGGL(ssd_scan, dim3(1), dim3(256), 0, stream, hist, state, shifts[p]);
  }

  hipLaunchKernelGGL(ssd_negsum, dim3(GRIDH_), dim3(256), 0, stream, keys, state, psum, pcnt);
  hipLaunchKernelGGL(ssd_final, dim3(1), dim3(256), 0, stream, psum, pcnt, state, scal, out);
}